// MRCNERModel_24979529793992
// MI455X (gfx1250) — compile-verified
//
#include <hip/hip_runtime.h>
#include <hip/hip_bf16.h>
#include <math.h>

// Problem dims (fixed by the reference)
#define BDIM 2
#define LDIM 256
#define HDIM 768
#define IDIM 1536
#define MROWS (BDIM * LDIM)   // 512 flattened (b,l) rows

typedef __attribute__((ext_vector_type(2))) float v2f;
typedef __attribute__((ext_vector_type(8))) float v8f;

// ---------------------------------------------------------------------------
// Kernel 1: a = X @ w1[:H] + b1 ; c = X @ w1[H:]     (fp32 WMMA 16x16x4)
//   X: (512, 768) row-major, W half: (768, 1536) row-major
//   blockIdx.z = 0 -> a (bias folded), 1 -> c
//   block = 256 threads = 8 waves arranged 2(M) x 4(N); block tile 32x64.
// ---------------------------------------------------------------------------
__global__ __launch_bounds__(256) void gemm_ac_kernel(
    const float* __restrict__ X,
    const float* __restrict__ W1,     // (2H, I)
    const float* __restrict__ b1,     // (I)
    float* __restrict__ ws)           // a at [0), c at [MROWS*IDIM)
{
    const int tid  = threadIdx.x;
    const int lane = tid & 31;
    const int wave = tid >> 5;
    const int wm   = wave >> 2;                 // 0..1
    const int wn   = wave & 3;                  // 0..3
    const int z    = blockIdx.z;

    const int mblock = blockIdx.y * 32;
    const int ncol   = blockIdx.x * 64 + wn * 16 + (lane & 15); // this lane's N column
    const int mloc   = wm * 16 + (lane & 15);                    // A-fragment row in LDS tile
    const int kh     = (lane >> 4) * 2;                          // K split by lane half

    const float* W = W1 + (size_t)z * HDIM * IDIM;               // select w1 half

    __shared__ float As[32][65];                                 // +1 pad

    v8f acc = {};
    for (int kc = 0; kc < HDIM; kc += 64) {
        // Cooperative, coalesced load of A tile: 32 rows x 64 k
        #pragma unroll
        for (int j = 0; j < 8; ++j) {
            int idx = tid + j * 256;
            int r   = idx >> 6;
            int cc  = idx & 63;
            As[r][cc] = X[(size_t)(mblock + r) * HDIM + kc + cc];
        }
        __syncthreads();

        #pragma unroll
        for (int k = 0; k < 64; k += 4) {
            // A fragment (16x4 fp32): lanes 0-15 K={k,k+1}, lanes 16-31 K={k+2,k+3}
            v2f af;
            af.x = As[mloc][k + kh];
            af.y = As[mloc][k + kh + 1];
            // B fragment (4x16 fp32): same K split, N = ncol (coalesced 16-lane rows)
            const float* wp = W + (size_t)(kc + k + kh) * IDIM + ncol;
            v2f bf;
            bf.x = wp[0];
            bf.y = wp[IDIM];
            acc = __builtin_amdgcn_wmma_f32_16x16x4_f32(
                false, af, false, bf, (short)0, acc, false, false);
        }
        __syncthreads();
    }

    // Epilogue: C/D layout -> VGPR j holds rows (j, j+8) split across lane halves
    float bias = (z == 0) ? b1[ncol] : 0.0f;
    float* out = ws + (size_t)z * MROWS * IDIM;
    const int mrow = mblock + wm * 16 + (lane >> 4) * 8;
    #pragma unroll
    for (int j = 0; j < 8; ++j) {
        out[(size_t)(mrow + j) * IDIM + ncol] = acc[j] + bias;
    }
}

// ---------------------------------------------------------------------------
// Kernel 2: start/end logits (one wave32 per (b,l) row)
// ---------------------------------------------------------------------------
__global__ __launch_bounds__(256) void logits_kernel(
    const float* __restrict__ hs,
    const float* __restrict__ sw, const float* __restrict__ sb,
    const float* __restrict__ ew, const float* __restrict__ eb,
    float* __restrict__ out)
{
    const int lane = threadIdx.x & 31;
    const int wave = threadIdx.x >> 5;
    const int row  = blockIdx.x * 8 + wave;          // 0..511
    const float* h = hs + (size_t)row * HDIM;

    float aS = 0.0f, aE = 0.0f;
    for (int k = lane; k < HDIM; k += 32) {
        float v = h[k];
        aS = fmaf(v, sw[k], aS);
        aE = fmaf(v, ew[k], aE);
    }
    #pragma unroll
    for (int off = 16; off > 0; off >>= 1) {
        aS += __shfl_down(aS, off, 32);
        aE += __shfl_down(aE, off, 32);
    }
    if (lane == 0) {
        out[row]          = aS + sb[0];
        out[MROWS + row]  = aE + eb[0];
    }
}

// ---------------------------------------------------------------------------
// Kernel 3: span[b,s,t] = sum_i gelu(a'[s,i] + c[t,i]) * w2[i] + b2
//   (b1 already folded into a'). 16x16 (s,t) tile per block, LDS-staged.
// ---------------------------------------------------------------------------
#define CHUNK 128
__global__ __launch_bounds__(256) void span_kernel(
    const float* __restrict__ ws,
    const float* __restrict__ w2,
    const float* __restrict__ b2,
    float* __restrict__ out)
{
    const float* A  = ws;                                // a + b1
    const float* Cm = ws + (size_t)MROWS * IDIM;         // c

    const int tid   = threadIdx.x;
    const int sl    = tid >> 4;
    const int tl    = tid & 15;
    const int b     = blockIdx.z;
    const int sbase = blockIdx.y * 16;
    const int tbase = blockIdx.x * 16;

    __shared__ float sa[16][CHUNK + 1];                  // +1 pad: bank-conflict free
    __shared__ float sc[16][CHUNK + 1];
    __shared__ float sw2[CHUNK];

    const float* Ab = A  + ((size_t)b * LDIM + sbase) * IDIM;
    const float* Cb = Cm + ((size_t)b * LDIM + tbase) * IDIM;

    float acc = 0.0f;
    for (int ic = 0; ic < IDIM; ic += CHUNK) {
        #pragma unroll
        for (int j = 0; j < (16 * CHUNK) / 256; ++j) {   // 8 iters, coalesced rows
            int idx = tid + j * 256;
            int r   = idx >> 7;                          // / CHUNK
            int cc  = idx & (CHUNK - 1);
            sa[r][cc] = Ab[(size_t)r * IDIM + ic + cc];
            sc[r][cc] = Cb[(size_t)r * IDIM + ic + cc];
        }
        if (tid < CHUNK) sw2[tid] = w2[ic + tid];
        __syncthreads();

        #pragma unroll 4
        for (int i = 0; i < CHUNK; ++i) {
            float x = sa[sl][i] + sc[tl][i];
            // exact GELU: 0.5*x*(1+erf(x/sqrt(2)))  (matches approximate=False)
            float g = 0.5f * x * (1.0f + erff(x * 0.70710678118654752440f));
            acc = fmaf(g, sw2[i], acc);
        }
        __syncthreads();
    }

    out[2 * MROWS + (size_t)b * LDIM * LDIM
        + (size_t)(sbase + sl) * LDIM + (tbase + tl)] = acc + b2[0];
}

// ---------------------------------------------------------------------------
extern "C" void kernel_launch(void* const* d_in, const int* in_sizes, int n_in,
                              void* d_out, int out_size, void* d_ws, size_t ws_size,
                              hipStream_t stream) {
    const float* hs      = (const float*)d_in[0];  // (B,L,H)
    const float* start_w = (const float*)d_in[1];  // (H)
    const float* start_b = (const float*)d_in[2];  // ()
    const float* end_w   = (const float*)d_in[3];  // (H)
    const float* end_b   = (const float*)d_in[4];  // ()
    const float* w1      = (const float*)d_in[5];  // (2H, I)
    const float* b1      = (const float*)d_in[6];  // (I)
    const float* w2      = (const float*)d_in[7];  // (I)
    const float* b2      = (const float*)d_in[8];  // ()
    float* out = (float*)d_out;
    float* ws  = (float*)d_ws;                     // needs 2*512*1536*4 = 6.3 MB

    // a' and c via fp32 WMMA GEMM
    gemm_ac_kernel<<<dim3(IDIM / 64, MROWS / 32, 2), 256, 0, stream>>>(hs, w1, b1, ws);
    // start/end logits
    logits_kernel<<<dim3(MROWS / 8), 256, 0, stream>>>(hs, start_w, start_b, end_w, end_b, out);
    // span logits (stream-ordered after GEMM)
    span_kernel<<<dim3(LDIM / 16, LDIM / 16, BDIM), 256, 0, stream>>>(ws, w2, b2, out);
}